// SVF_2D_12103217840715
// MI455X (gfx1250) — compile-verified
//
#include <hip/hip_runtime.h>

// SVF scaling-and-squaring, MI455X (gfx1250).
// Gather/bandwidth-bound: no matrix math -> no WMMA; optimized for the memory
// system instead: float2-interleaved field (one b64 gather fetches both
// channels), field buffer L2-resident (105 MB vs 192 MB L2), ASYNCcnt
// global->LDS staging of the identity grid, NT hints on stream-once data.
// Steps ping-pong half1(d_out) <-> d_ws (or half2 of d_out if ws is small);
// all passes read/write disjoint regions -> race-free and deterministic.

constexpr int Nb = 32;
constexpr int Hh = 640;
constexpr int Ww = 640;
constexpr int HW = Hh * Ww;            // 409600 = 1600 * 256 (exact)
constexpr int NHW = Nb * HW;           // pixels total (each pixel = float2)
constexpr int BLOCK_X = 320;           // 10 waves; 2 blocks cover a row exactly
constexpr int NSTEPS = 32;

// ---------------------------------------------------------------- init ------
// v [N,2,H,W] f32 -> interleaved float2 field, scaled by 2^-32 (exact).
// grid (HW/256, Nb): division-free indexing.
__global__ void svf_init(const float* __restrict__ v, float2* __restrict__ dst) {
  const int s = blockIdx.x * blockDim.x + threadIdx.x;   // pixel within image
  const int n = blockIdx.y;                              // batch
  const float* vp = v + (size_t)(2 * n) * HW + s;
  float a = __builtin_nontemporal_load(vp);              // read-once stream
  float b = __builtin_nontemporal_load(vp + HW);
  dst[(size_t)n * HW + s] = make_float2(a * 0x1p-32f, b * 0x1p-32f);
}

// ---------------------------------------------------- bilinear (border) -----
__device__ __forceinline__ float2 bilinear_border(const float2* __restrict__ f,
                                                  float sx, float sy) {
  sx = fminf(fmaxf(sx, 0.0f), (float)(Ww - 1));          // padding_mode='border'
  sy = fminf(fmaxf(sy, 0.0f), (float)(Hh - 1));
  float x0f = floorf(sx), y0f = floorf(sy);
  int x0 = (int)x0f, y0 = (int)y0f;
  int x1 = min(x0 + 1, Ww - 1);
  int y1 = min(y0 + 1, Hh - 1);
  float wx = sx - x0f, wy = sy - y0f;
  // 4 x b64 gathers; displacement field is smooth -> adjacent lanes hit
  // adjacent cachelines; whole source buffer stays L2-resident.
  float2 v00 = f[y0 * Ww + x0];
  float2 v01 = f[y0 * Ww + x1];
  float2 v10 = f[y1 * Ww + x0];
  float2 v11 = f[y1 * Ww + x1];
  float tx = v00.x + wx * (v01.x - v00.x);
  float ty = v00.y + wx * (v01.y - v00.y);
  float bx = v10.x + wx * (v11.x - v10.x);
  float by = v10.y + wx * (v11.y - v10.y);
  return make_float2(tx + wy * (bx - tx), ty + wy * (by - ty));
}

// CDNA5 ASYNCcnt path: async-stage this block's identity-grid row segment
// (shared across all 32 batches, L2-resident) into LDS, overlap with the
// coalesced load of the lane's own displacement, then wait on ASYNCcnt.
// Each lane reads only its own staged element -> no cross-wave barrier needed.
__device__ __forceinline__ float2 stage_ig(const float2* __restrict__ ig,
                                           float2* s_ig, int x, int y) {
  unsigned lds = (unsigned)(size_t)(&s_ig[threadIdx.x]);  // low 32b = LDS addr
  unsigned long long ga =
      (unsigned long long)(const void*)(ig + (size_t)y * Ww + x);
  asm volatile("global_load_async_to_lds_b64 %0, %1, off"
               :: "v"(lds), "v"(ga) : "memory");
  return make_float2(0.f, 0.f);  // value read later, after the wait
}

__device__ __forceinline__ void wait_async0() {
#if __has_builtin(__builtin_amdgcn_s_wait_asynccnt)
  __builtin_amdgcn_s_wait_asynccnt(0);
#else
  asm volatile("s_wait_asynccnt 0x0" ::: "memory");
#endif
}

// ------------------------------------------------------------- one step -----
// out[p] = in[p] + sample(in, identity[p] + in[p])
__global__ __launch_bounds__(BLOCK_X)
void svf_step(const float2* __restrict__ in, float2* __restrict__ out,
              const float2* __restrict__ ig) {
  __shared__ float2 s_ig[BLOCK_X];
  const int x = blockIdx.x * BLOCK_X + threadIdx.x;       // exact: 2*320 == 640
  const int y = blockIdx.y;
  const int n = blockIdx.z;

  stage_ig(ig, s_ig, x, y);

  const float2* __restrict__ f = in + (size_t)n * HW;
  float2 cur = f[y * Ww + x];                             // overlaps async latency

  wait_async0();
  float2 g = s_ig[threadIdx.x];

  // unnormalize, align_corners=True: s = (g + d + 1) * 0.5 * (dim-1)
  float sx = fmaf(g.x + cur.x, 0.5f * (Ww - 1), 0.5f * (Ww - 1));
  float sy = fmaf(g.y + cur.y, 0.5f * (Hh - 1), 0.5f * (Hh - 1));
  float2 s = bilinear_border(f, sx, sy);

  out[(size_t)n * HW + y * Ww + x] = make_float2(cur.x + s.x, cur.y + s.y);
}

// ------------------------------------------- fused final step (ws path) -----
// Performs the 32nd squaring step AND writes both NCHW outputs directly.
// Reads only from `in` (= d_ws) -> disjoint from d_out, race-free.
__global__ __launch_bounds__(BLOCK_X)
void svf_final_fused(const float2* __restrict__ in, const float2* __restrict__ ig,
                     float* __restrict__ transT, float* __restrict__ dispD) {
  __shared__ float2 s_ig[BLOCK_X];
  const int x = blockIdx.x * BLOCK_X + threadIdx.x;
  const int y = blockIdx.y;
  const int n = blockIdx.z;

  stage_ig(ig, s_ig, x, y);

  const float2* __restrict__ f = in + (size_t)n * HW;
  float2 cur = f[y * Ww + x];

  wait_async0();
  float2 g = s_ig[threadIdx.x];

  float sx = fmaf(g.x + cur.x, 0.5f * (Ww - 1), 0.5f * (Ww - 1));
  float sy = fmaf(g.y + cur.y, 0.5f * (Hh - 1), 0.5f * (Hh - 1));
  float2 s = bilinear_border(f, sx, sy);

  float dx = cur.x + s.x, dy = cur.y + s.y;               // final displacement
  size_t p0 = (size_t)(2 * n) * HW + (size_t)y * Ww + x;  // channel 0 plane
  size_t p1 = p0 + HW;                                    // channel 1 plane
  __builtin_nontemporal_store(dx, dispD + p0);            // write-once streams
  __builtin_nontemporal_store(dy, dispD + p1);
  __builtin_nontemporal_store(g.x + dx, transT + p0);     // transformation
  __builtin_nontemporal_store(g.y + dy, transT + p1);
}

// -------------------------------- zero-workspace fallback formatting --------
// Pass 1: interleaved field (half1) -> NCHW displacement (half2).
__global__ void svf_deinterleave(const float2* __restrict__ a,
                                 float* __restrict__ dispD) {
  const int s = blockIdx.x * blockDim.x + threadIdx.x;
  const int n = blockIdx.y;
  float2 d = a[(size_t)n * HW + s];
  __builtin_nontemporal_store(d.x, dispD + (size_t)(2 * n) * HW + s);
  __builtin_nontemporal_store(d.y, dispD + (size_t)(2 * n + 1) * HW + s);
}

// Pass 2: NCHW displacement (half2) -> transformation = disp + identity (half1).
__global__ void svf_transform(const float* __restrict__ dispD,
                              const float2* __restrict__ ig,
                              float* __restrict__ transT) {
  const int s = blockIdx.x * blockDim.x + threadIdx.x;
  const int n = blockIdx.y;
  float2 g = ig[s];
  size_t p0 = (size_t)(2 * n) * HW + s;
  size_t p1 = p0 + HW;
  __builtin_nontemporal_store(g.x + dispD[p0], transT + p0);
  __builtin_nontemporal_store(g.y + dispD[p1], transT + p1);
}

// ---------------------------------------------------------------------------
extern "C" void kernel_launch(void* const* d_in, const int* in_sizes, int n_in,
                              void* d_out, int out_size, void* d_ws,
                              size_t ws_size, hipStream_t stream) {
  (void)in_sizes; (void)n_in; (void)out_size;

  const float*  v  = (const float*)d_in[0];    // [32,2,640,640] f32
  const float2* ig = (const float2*)d_in[1];   // [1,640,640,2]  f32 (xy-interleaved)

  float*  out  = (float*)d_out;                // [transformation | displacement]
  float2* bufA = (float2*)d_out;               // half1 as interleaved scratch
  float*  dispOut = out + (size_t)Nb * 2 * HW; // half2 (displacement slot)

  dim3 flatGrid(HW / 256, Nb);                 // (1600, 32), division-free
  dim3 stepGrid(Ww / BLOCK_X, Hh, Nb);         // (2, 640, 32)
  dim3 stepBlock(BLOCK_X);

  // disp = v / 2^32, interleaved, into half1
  svf_init<<<flatGrid, 256, 0, stream>>>(v, bufA);

  const size_t fieldBytes = (size_t)NHW * sizeof(float2);  // 104,857,600 B
  if (ws_size >= fieldBytes) {
    // Preferred: ping-pong half1 <-> ws; step 30 (even k reads A) lands in ws;
    // fused final step reads ws and writes both NCHW outputs directly.
    float2* bufW = (float2*)d_ws;
    for (int k = 0; k < NSTEPS - 1; ++k) {     // 31 steps
      const float2* src = (k & 1) ? bufW : bufA;
      float2*       dst = (k & 1) ? bufA : bufW;
      svf_step<<<stepGrid, stepBlock, 0, stream>>>(src, dst, ig);
    }
    svf_final_fused<<<stepGrid, stepBlock, 0, stream>>>(bufW, ig, out, dispOut);
  } else {
    // Fallback: ping-pong half1 <-> half2; 32 steps end in half1; then two
    // formatting passes, each touching disjoint halves.
    float2* bufB = (float2*)dispOut;
    for (int k = 0; k < NSTEPS; ++k) {
      const float2* src = (k & 1) ? bufB : bufA;
      float2*       dst = (k & 1) ? bufA : bufB;
      svf_step<<<stepGrid, stepBlock, 0, stream>>>(src, dst, ig);
    }
    svf_deinterleave<<<flatGrid, 256, 0, stream>>>(bufA, dispOut);
    svf_transform<<<flatGrid, 256, 0, stream>>>(dispOut, ig, out);
  }
}